// GCN_20675972563377
// MI455X (gfx1250) — compile-verified
//
#include <hip/hip_runtime.h>

typedef float v2f __attribute__((ext_vector_type(2)));
typedef float v8f __attribute__((ext_vector_type(8)));

// ---------------------------------------------------------------------------
// Elementwise fill
// ---------------------------------------------------------------------------
__global__ __launch_bounds__(256) void k_fill(float* __restrict__ p, float v, long n) {
  long i = (long)blockIdx.x * blockDim.x + threadIdx.x;
  if (i < n) p[i] = v;
}

// ---------------------------------------------------------------------------
// Degree accumulation over destination nodes (self loop pre-seeded as 1.0)
// ---------------------------------------------------------------------------
__global__ __launch_bounds__(256) void k_deg(const int* __restrict__ dst,
                                             float* __restrict__ deg, int E) {
  int i = blockIdx.x * blockDim.x + threadIdx.x;
  if (i < E) unsafeAtomicAdd(&deg[dst[i]], 1.0f);
}

__global__ __launch_bounds__(256) void k_rsqrt(float* __restrict__ d, int n) {
  int i = blockIdx.x * blockDim.x + threadIdx.x;
  if (i < n) d[i] = rsqrtf(d[i]);   // deg >= 1 always (self loop)
}

// ---------------------------------------------------------------------------
// H = A(MxK=128) * W(128xOUT) via V_WMMA_F32_16X16X4_F32, one 16-row strip
// per wave32, all OUT columns held in v8f accumulators.
//
// A-frag (16x4 f32): lane = m (0..15) | m+16; VGPRs = A[m][k + 2*kh + {0,1}]
// B-frag (4x16 f32): lane = n | n+16;        VGPRs = W[k + 2*kh + {0,1}][n]
// C/D (16x16 f32):   VGPR r holds row (r + 8*kh), col = n
// ---------------------------------------------------------------------------
template <int OUT>
__global__ __launch_bounds__(256)
void k_gemm(const float* __restrict__ A, const float* __restrict__ W,
            float* __restrict__ H, int M) {
  constexpr int NT = OUT / 16;
  int wave = (blockIdx.x * blockDim.x + threadIdx.x) >> 5;
  int lane = threadIdx.x & 31;
  int row0 = wave * 16;
  if (row0 >= M) return;                 // wave-uniform: EXEC stays all-ones

  int m  = lane & 15;                    // A row within strip / B column
  int kh = lane >> 4;                    // K-half selector

  v8f acc[NT] = {};

  int arow = (row0 + m < M) ? (row0 + m) : (M - 1);   // clamp for ragged tail
  const float* ap = A + (size_t)arow * 128 + 2 * kh;

  for (int k = 0; k < 128; k += 4) {
    v2f a = *(const v2f*)(ap + k);                     // 8B aligned load
    const float* bp = W + (size_t)(k + 2 * kh) * OUT + m;
#pragma unroll
    for (int t = 0; t < NT; ++t) {
      v2f b;
      b.x = bp[t * 16];
      b.y = bp[t * 16 + OUT];
      acc[t] = __builtin_amdgcn_wmma_f32_16x16x4_f32(
          /*neg_a=*/false, a, /*neg_b=*/false, b,
          /*c_mod=*/(short)0, acc[t], /*reuse_a=*/false, /*reuse_b=*/false);
    }
  }

  // Write back: lane half kh owns rows row0+8*kh .. row0+8*kh+7, column m+16t.
  int rbase = row0 + 8 * kh;
#pragma unroll
  for (int t = 0; t < NT; ++t) {
#pragma unroll
    for (int r = 0; r < 8; ++r) {
      int row = rbase + r;
      if (row < M) H[(size_t)row * OUT + t * 16 + m] = acc[t][r];
    }
  }
}

// ---------------------------------------------------------------------------
// Edge scatter: one wave32 per edge; lane k covers dims k, k+32, ...
// agg[dst,:] += h[src,:] * dinv[src] * dinv[dst]   (hardware f32 atomics)
// ---------------------------------------------------------------------------
template <int OUT>
__global__ __launch_bounds__(256)
void k_scatter(const int* __restrict__ src, const int* __restrict__ dst,
               const float* __restrict__ dinv, const float* __restrict__ h,
               float* __restrict__ agg, int E) {
  int wave = (blockIdx.x * blockDim.x + threadIdx.x) >> 5;
  int lane = threadIdx.x & 31;
  if (wave >= E) return;
  int s = src[wave];
  int d = dst[wave];
  float nrm = dinv[s] * dinv[d];
  const float* hs = h + (size_t)s * OUT;
  float* ad = agg + (size_t)d * OUT;
#pragma unroll
  for (int k = lane; k < OUT; k += 32)
    unsafeAtomicAdd(ad + k, hs[k] * nrm);
}

// ---------------------------------------------------------------------------
// Self-loop term + bias + optional ReLU, in place on agg.
// out[i,d] = act(agg[i,d] + h[i,d]*dinv[i]^2 + b[d])
// ---------------------------------------------------------------------------
template <int OUT, bool RELU>
__global__ __launch_bounds__(256)
void k_act(float* __restrict__ agg, const float* __restrict__ h,
           const float* __restrict__ dinv, const float* __restrict__ b, int N) {
  long i = (long)blockIdx.x * blockDim.x + threadIdx.x;
  long tot = (long)N * OUT;
  if (i >= tot) return;
  int node = (int)(i / OUT);
  int dcol = (int)(i % OUT);
  float di = dinv[node];
  float v = agg[i] + h[i] * di * di + b[dcol];
  agg[i] = RELU ? fmaxf(v, 0.0f) : v;
}

// ---------------------------------------------------------------------------
static inline int cdiv(long a, long b) { return (int)((a + b - 1) / b); }

extern "C" void kernel_launch(void* const* d_in, const int* in_sizes, int n_in,
                              void* d_out, int out_size, void* d_ws, size_t ws_size,
                              hipStream_t stream) {
  const float* x  = (const float*)d_in[0];
  const int*   ei = (const int*)d_in[1];
  const float* W1 = (const float*)d_in[2];
  const float* b1 = (const float*)d_in[3];
  const float* W2 = (const float*)d_in[4];
  const float* b2 = (const float*)d_in[5];
  const float* W3 = (const float*)d_in[6];
  const float* b3 = (const float*)d_in[7];

  const int N = in_sizes[0] / 128;
  const int E = in_sizes[1] / 2;
  const int* src = ei;
  const int* dst = ei + E;

  char* ws = (char*)d_ws;
  size_t off0 = 0;
  size_t off1 = (((size_t)N * 4) + 255) & ~(size_t)255;       // after dinv
  size_t off2 = off1 + (size_t)N * 128 * 4;                   // after bufH
  float* dinv = (float*)(ws + off0);
  float* bufH = (float*)(ws + off1);   // h = feat @ W
  float* bufF = (float*)(ws + off2);   // aggregated features / next-layer input
  float* out  = (float*)d_out;

  const int T = 256;
  const int gemmBlocks = cdiv((long)cdiv(N, 16) * 32, T);
  const int edgeBlocks = cdiv((long)E * 32, T);

  // --- normalization: deg (with self loop) -> d^-1/2 ---
  k_fill<<<cdiv(N, T), T, 0, stream>>>(dinv, 1.0f, N);
  k_deg<<<cdiv(E, T), T, 0, stream>>>(dst, dinv, E);
  k_rsqrt<<<cdiv(N, T), T, 0, stream>>>(dinv, N);

  // --- layer 1: x -> bufF (ReLU) ---
  k_gemm<128><<<gemmBlocks, T, 0, stream>>>(x, W1, bufH, N);
  k_fill<<<cdiv((long)N * 128, T), T, 0, stream>>>(bufF, 0.0f, (long)N * 128);
  k_scatter<128><<<edgeBlocks, T, 0, stream>>>(src, dst, dinv, bufH, bufF, E);
  k_act<128, true><<<cdiv((long)N * 128, T), T, 0, stream>>>(bufF, bufH, dinv, b1, N);

  // --- layer 2: bufF -> bufF (ReLU); gemm consumes bufF before it is zeroed ---
  k_gemm<128><<<gemmBlocks, T, 0, stream>>>(bufF, W2, bufH, N);
  k_fill<<<cdiv((long)N * 128, T), T, 0, stream>>>(bufF, 0.0f, (long)N * 128);
  k_scatter<128><<<edgeBlocks, T, 0, stream>>>(src, dst, dinv, bufH, bufF, E);
  k_act<128, true><<<cdiv((long)N * 128, T), T, 0, stream>>>(bufF, bufH, dinv, b2, N);

  // --- layer 3: bufF -> d_out (no ReLU), aggregate directly into d_out ---
  k_gemm<64><<<gemmBlocks, T, 0, stream>>>(bufF, W3, bufH, N);
  k_fill<<<cdiv((long)N * 64, T), T, 0, stream>>>(out, 0.0f, (long)N * 64);
  k_scatter<64><<<edgeBlocks, T, 0, stream>>>(src, dst, dinv, bufH, out, E);
  k_act<64, false><<<cdiv((long)N * 64, T), T, 0, stream>>>(out, bufH, dinv, b3, N);
}